// CompoundEmbedding_79989470921233
// MI455X (gfx1250) — compile-verified
//
#include <hip/hip_runtime.h>

// Embedding-bag SUM, tuned for MI455X (gfx1250, wave32).
//   weight: [1,000,000, 64] f32   input: [16384, 50] int   out: [16384, 64] f32
// Memory-bound random gather (~220 MB @ 23.3 TB/s => ~9.4 us floor). WMMA does
// not apply (per-bag row sets differ -> no shared B operand). Wins:
//   - half-wave per bag: each wave-wide global_load_b128 moves two full 256B rows
//   - 32-bit saddr-form addressing (1 VALU op per gather instead of 64-bit mads)
//   - 10 in-flight b128 loads per wave for deep MLP against HBM latency
//   - CDNA5 async global->LDS staging of the index block + s_wait_asynccnt

#define EMB_DIM        64
#define BAG            50
#define BAGS_PER_BLOCK 16
#define THREADS        256   // 8 wave32 waves; half-wave (16 lanes) per bag row

__global__ __launch_bounds__(THREADS)
void embbag_sum_kernel(const int* __restrict__ input,    // [B, BAG] int32 indices
                       const float* __restrict__ weight, // [1e6, 64] f32
                       float* __restrict__ out)          // [B, 64] f32
{
    // 16 bags * 50 idx * 4B = 3200 B of indices for this block.
    __shared__ __align__(16) int sIdx[BAGS_PER_BLOCK * BAG];

    const int t       = threadIdx.x;
    const int bagBase = blockIdx.x * BAGS_PER_BLOCK;

    // ---- Stage indices global -> LDS with CDNA5 async copies (b128 = 4 ints) ----
    {
        constexpr int chunks = (BAGS_PER_BLOCK * BAG * 4) / 16; // 200 x 16B
        if (t < chunks) {
            const unsigned lds_off = (unsigned)(size_t)(&sIdx[t * 4]);
            const unsigned long long gaddr =
                (unsigned long long)(size_t)(input + (size_t)bagBase * BAG + (size_t)t * 4);
            asm volatile("global_load_async_to_lds_b128 %0, %1, off"
                         :: "v"(lds_off), "v"(gaddr) : "memory");
        }
        asm volatile("s_wait_asynccnt 0x0" ::: "memory");
    }
    __syncthreads();

    // ---- Gather + accumulate: one bag per half-wave, float4 per lane ----
    const int bagLocal = t >> 4;                 // 0..15
    const unsigned dimByte = (t & 15) * 16u;     // byte offset of this lane's float4
    const int* __restrict__ bagIdx = &sIdx[bagLocal * BAG];
    const char* __restrict__ wbytes = (const char*)weight;

    float accx = 0.f, accy = 0.f, accz = 0.f, accw = 0.f;

    #pragma unroll
    for (int k0 = 0; k0 < BAG; k0 += 10) {       // 5 groups of 10 in-flight b128 loads
        float4 r[10];
        #pragma unroll
        for (int j = 0; j < 10; ++j) {
            unsigned row = (unsigned)bagIdx[k0 + j];        // LDS broadcast read
            __builtin_assume(row < (1u << 20));             // 1e6 rows -> 24-bit mad ok
            const unsigned byteOff = row * 256u + dimByte;  // < 2^28, fits u32
            // base(SGPR pair) + zext(u32 voffset) -> GVS saddr-form global_load_b128
            r[j] = *(const float4*)(wbytes + (size_t)byteOff);
        }
        #pragma unroll
        for (int j = 0; j < 10; ++j) {
            accx += r[j].x; accy += r[j].y; accz += r[j].z; accw += r[j].w;
        }
    }

    float4 res = make_float4(accx, accy, accz, accw);
    char* obytes = (char*)out;
    const unsigned outOff = (unsigned)(bagBase + bagLocal) * 256u + dimByte;
    *(float4*)(obytes + (size_t)outOff) = res;
}

extern "C" void kernel_launch(void* const* d_in, const int* in_sizes, int n_in,
                              void* d_out, int out_size, void* d_ws, size_t ws_size,
                              hipStream_t stream)
{
    (void)in_sizes; (void)n_in; (void)d_ws; (void)ws_size;

    const int*   input  = (const int*)d_in[0];    // setup_inputs order: "input", "weight"
    const float* weight = (const float*)d_in[1];
    float*       out    = (float*)d_out;

    const int batch  = out_size / EMB_DIM;               // 16384
    const int blocks = batch / BAGS_PER_BLOCK;           // 1024

    embbag_sum_kernel<<<blocks, THREADS, 0, stream>>>(input, weight, out);
}